// SpikeFP32SiLU_69982197121560
// MI455X (gfx1250) — compile-verified
//
#include <hip/hip_runtime.h>
#include <cstdint>

#ifndef __has_builtin
#define __has_builtin(x) 0
#endif

typedef __attribute__((ext_vector_type(4))) unsigned int tdm_v4u;
typedef __attribute__((ext_vector_type(4))) int          tdm_v4i;
typedef __attribute__((ext_vector_type(8))) int          tdm_v8i;
typedef __attribute__((ext_vector_type(4))) float        f32x4;

#define HAVE_TDM __has_builtin(__builtin_amdgcn_tensor_load_to_lds)

constexpr int kVals      = 256;              // values per tile == blockDim.x
constexpr int kBits      = 32;               // bit planes per value
constexpr int kTileElems = kVals * kBits;    // 8192 floats = 32 KB per LDS buffer

__device__ __forceinline__ uint32_t clamp_u32(long long x) {
  return (x > 0xFFFFFFFFLL) ? 0xFFFFFFFFu : (uint32_t)x;
}

#if HAVE_TDM
// Issue one TDM DMA: contiguous 1-D tile of kTileElems f32 from global -> LDS.
// D# layout per CDNA5 ISA 8.3/8.4:
//  g0: [1:0]=count=1, [63:32]=lds_addr, [120:64]=global_addr, [127:126]=type=2
//  g1: data_size=4B (code 2), tensor_dim0=remaining elems (OOB reads -> 0),
//      tensor_dim1=1, tile_dim0=kTileElems, tile_dim1/2=0 (unused),
//      tensor_dim0_stride=kTileElems (benign, Y loop unused)
__device__ __forceinline__ void tdm_issue(const float* gsrc, uint32_t lds_off,
                                          uint32_t dim0_elems) {
  const uint64_t ga = (uint64_t)(uintptr_t)gsrc;
  tdm_v4u g0 = { 1u,                                     // count=1, no gather
                 lds_off,                                // LDS byte address
                 (uint32_t)(ga & 0xFFFFFFFFu),           // global_addr[31:0]
                 (uint32_t)((ga >> 32) & 0x01FFFFFFu) |  // global_addr[56:32]
                     0x80000000u };                      // type=2 ("image")
  tdm_v8i g1 = {
      (int)(2u << 16),                                   // data_size = 4 bytes
      (int)((dim0_elems & 0xFFFFu) << 16),               // tensor_dim0[15:0]
      (int)(((dim0_elems >> 16) & 0xFFFFu) | (1u << 16)),// dim0[31:16] | tensor_dim1=1
      (int)((uint32_t)kTileElems << 16),                 // tensor_dim1 hi=0 | tile_dim0
      0,                                                 // tile_dim1=0, tile_dim2=0
      kTileElems,                                        // tensor_dim0_stride lo
      0, 0 };
  tdm_v4i z4 = { 0, 0, 0, 0 };
#if __clang_major__ >= 23
  tdm_v8i z8 = { 0, 0, 0, 0, 0, 0, 0, 0 };
  __builtin_amdgcn_tensor_load_to_lds(g0, g1, z4, z4, z8, 0);
#else
  __builtin_amdgcn_tensor_load_to_lds(g0, g1, z4, z4, 0);
#endif
}
#endif  // HAVE_TDM

__global__ void __launch_bounds__(kVals)
spike_silu_tdm(const float* __restrict__ in, float* __restrict__ out,
               long long n_vals, long long ntiles, long long tpb) {
#if HAVE_TDM
  __shared__ float lds[2][kTileElems];                   // 64 KB, double buffered
#endif
  const long long total_f = n_vals * (long long)kBits;
  long long k0 = (long long)blockIdx.x * tpb;
  if (k0 >= ntiles) return;
  long long k1 = k0 + tpb;
  if (k1 > ntiles) k1 = ntiles;

  const int tid  = (int)threadIdx.x;
  const int lane = tid & 31;

#if HAVE_TDM
  // TDM ignores EXEC, so gate issuance with a wave-uniform (scalar) branch.
  const bool issuer =
      (__builtin_amdgcn_readfirstlane((int)(threadIdx.x >> 5)) == 0);
  const uint32_t lds_off0 = (uint32_t)(uintptr_t)(&lds[0][0]);
  const uint32_t lds_off1 = (uint32_t)(uintptr_t)(&lds[1][0]);

  if (issuer) {                                          // prologue: tile k0
    const long long e0 = k0 * (long long)kTileElems;
    tdm_issue(in + e0, lds_off0, clamp_u32(total_f - e0));
  }
#endif

  for (long long k = k0; k < k1; ++k) {
    const int buf = (int)((k - k0) & 1);
#if HAVE_TDM
    if (issuer) {
      if (k + 1 < k1) {                                  // prefetch next tile
        const long long e = (k + 1) * (long long)kTileElems;
        tdm_issue(in + e, (buf ? lds_off0 : lds_off1), clamp_u32(total_f - e));
        __builtin_amdgcn_s_wait_tensorcnt(1);            // tile k landed
      } else {
        __builtin_amdgcn_s_wait_tensorcnt(0);            // drain
      }
    }
    __syncthreads();                                     // tile k visible to all
#endif
    const long long vid = k * (long long)kVals + tid;
    if (vid < n_vals) {
      uint32_t u;
#if HAVE_TDM
      // Rotated read order -> every lane hits a distinct LDS bank per load.
      const float* base = &lds[buf][tid << 5];
      uint32_t r = 0;
#pragma unroll
      for (int j = 0; j < kBits; ++j) {
        const int idx = (j + lane) & 31;
        r = (r << 1) | ((__float_as_uint(base[idx]) >> 23) & 1u);
      }
      u = (r >> lane) | (r << ((32 - lane) & 31));       // rotr(r, lane)
#else
      const float* base = in + vid * (long long)kBits;
      u = 0;
#pragma unroll
      for (int j = 0; j < kBits; ++j)
        u = (u << 1) | ((__float_as_uint(base[j]) >> 23) & 1u);
#endif
      const float v  = __uint_as_float(u);
      const float ex = __expf(-v);
#if __has_builtin(__builtin_amdgcn_rcpf)
      const float y = v * __builtin_amdgcn_rcpf(1.0f + ex);
#else
      const float y = v / (1.0f + ex);
#endif
      const uint32_t yu = __float_as_uint(y);
      float* op = out + vid * (long long)kBits;
#pragma unroll
      for (int q = 0; q < 8; ++q) {
        f32x4 o = { (float)((yu >> (31 - (4 * q + 0))) & 1u),
                    (float)((yu >> (31 - (4 * q + 1))) & 1u),
                    (float)((yu >> (31 - (4 * q + 2))) & 1u),
                    (float)((yu >> (31 - (4 * q + 3))) & 1u) };
        __builtin_nontemporal_store(o, (f32x4*)op + q);  // stream 1 GB past L2
      }
    }
#if HAVE_TDM
    __syncthreads();  // all readers done before buf is DMA-overwritten (k+2)
#endif
  }
}

extern "C" void kernel_launch(void* const* d_in, const int* in_sizes, int n_in,
                              void* d_out, int out_size, void* d_ws, size_t ws_size,
                              hipStream_t stream) {
  (void)n_in; (void)out_size; (void)d_ws; (void)ws_size;
  const float* in = (const float*)d_in[0];
  float* out = (float*)d_out;

  const long long total_f = (long long)in_sizes[0];      // 134,217,728
  const long long n_vals  = total_f / kBits;             // 4,194,304
  if (n_vals <= 0) return;
  const long long ntiles = (n_vals + kVals - 1) / kVals; // 16,384

  long long nblocks = ntiles < 4096 ? ntiles : 4096;     // ~4096 persistent-ish blocks
  const long long tpb = (ntiles + nblocks - 1) / nblocks;
  nblocks = (ntiles + tpb - 1) / tpb;

  spike_silu_tdm<<<(dim3)(unsigned)nblocks, kVals, 0, stream>>>(
      in, out, n_vals, ntiles, tpb);
}